// GCN2_67095979098487
// MI455X (gfx1250) — compile-verified
//
#include <hip/hip_runtime.h>

#define N_NODES 100000
#define N_EDGES 1600000
#define DIM     128
#define HID     16
#define NCLS    40
#define NLAYERS 3
#define BN_EPS  1e-5f

typedef __attribute__((ext_vector_type(2))) float v2f;
typedef __attribute__((ext_vector_type(8))) float v8f;

// ---------------------------------------------------------------------------
// CSR row_ptr from sorted COO rows: row_ptr[r] = first edge with row[e] >= r
// ---------------------------------------------------------------------------
__global__ void build_row_ptr_kernel(const int* __restrict__ row,
                                     int* __restrict__ row_ptr, int E, int n) {
    int e = blockIdx.x * blockDim.x + threadIdx.x;
    if (e >= E) return;
    int r = row[e];
    int prev = (e == 0) ? -1 : row[e - 1];
    for (int q = prev + 1; q <= r; ++q) row_ptr[q] = e;
    if (e == E - 1) {
        for (int q = r + 1; q <= n; ++q) row_ptr[q] = E;
    }
}

// ---------------------------------------------------------------------------
// C[n x 16] = X[n x 128] @ W[128 x 16]   (fp32 WMMA 16x16x4, K looped by 4)
// One wave -> one 16-node tile. 8 waves / block.
// ---------------------------------------------------------------------------
__global__ void gemm_d2h_wmma(const float* __restrict__ X,
                              const float* __restrict__ W,
                              float* __restrict__ C, int ntiles) {
    int wave = threadIdx.x >> 5;
    int tile = blockIdx.x * (blockDim.x >> 5) + wave;
    if (tile >= ntiles) return;                 // wave-uniform: EXEC stays full
    int lane = threadIdx.x & 31;
    int r  = lane & 15;                         // A row / B col / D col
    int hs = lane >> 4;                         // half-wave select
    int node0 = tile * 16;

    v8f acc = {};
    const float* xrow = X + (size_t)(node0 + r) * DIM + 2 * hs;
#pragma unroll 4
    for (int k = 0; k < DIM; k += 4) {
        v2f a = *reinterpret_cast<const v2f*>(xrow + k);   // A[r][k+2hs .. +1]
        int kk = k + 2 * hs;
        v2f b;
        b.x = W[kk * HID + r];                  // B[k+2hs][r]
        b.y = W[(kk + 1) * HID + r];            // B[k+2hs+1][r]
        acc = __builtin_amdgcn_wmma_f32_16x16x4_f32(false, a, false, b,
                                                    (short)0, acc, false, false);
    }
#pragma unroll
    for (int v = 0; v < 8; ++v) {
        int m = v + 8 * hs;                     // D row
        C[(size_t)(node0 + m) * HID + r] = acc[v];
    }
}

// ---------------------------------------------------------------------------
// C[n x 128] = Hs[n x 16] @ W[16 x 128]
// One block -> one 16-node tile; wave w handles column tile w (8 waves).
// ---------------------------------------------------------------------------
__global__ void gemm_h2d_wmma(const float* __restrict__ Hs,
                              const float* __restrict__ W,
                              float* __restrict__ C, int ntiles) {
    int wave = threadIdx.x >> 5;
    int tile = blockIdx.x;
    if (tile >= ntiles) return;
    int lane = threadIdx.x & 31;
    int r  = lane & 15;
    int hs = lane >> 4;
    int node0 = tile * 16;
    int col0  = wave * 16;

    v8f acc = {};
    const float* hrow = Hs + (size_t)(node0 + r) * HID + 2 * hs;
#pragma unroll
    for (int k = 0; k < HID; k += 4) {
        v2f a = *reinterpret_cast<const v2f*>(hrow + k);
        int kk = k + 2 * hs;
        v2f b;
        b.x = W[kk * DIM + col0 + r];
        b.y = W[(kk + 1) * DIM + col0 + r];
        acc = __builtin_amdgcn_wmma_f32_16x16x4_f32(false, a, false, b,
                                                    (short)0, acc, false, false);
    }
#pragma unroll
    for (int v = 0; v < 8; ++v) {
        int m = v + 8 * hs;
        C[(size_t)(node0 + m) * DIM + col0 + r] = acc[v];
    }
}

// ---------------------------------------------------------------------------
// C[n x 40] = Hs[n x 16] @ W[16 x 40]   (3 column tiles, last partially valid)
// blockDim = 96 (3 waves). B loads clamped + mask-multiplied: branch-free so
// EXEC remains all-ones around the WMMAs.
// ---------------------------------------------------------------------------
__global__ void gemm_h2c_wmma(const float* __restrict__ Hs,
                              const float* __restrict__ W,
                              float* __restrict__ C, int ntiles) {
    int wave = threadIdx.x >> 5;
    int tile = blockIdx.x;
    if (tile >= ntiles) return;
    int lane = threadIdx.x & 31;
    int r  = lane & 15;
    int hs = lane >> 4;
    int node0 = tile * 16;
    int col0  = wave * 16;
    int col   = col0 + r;
    int colc  = (col < NCLS) ? col : (NCLS - 1);
    float msk = (col < NCLS) ? 1.0f : 0.0f;

    v8f acc = {};
    const float* hrow = Hs + (size_t)(node0 + r) * HID + 2 * hs;
#pragma unroll
    for (int k = 0; k < HID; k += 4) {
        v2f a = *reinterpret_cast<const v2f*>(hrow + k);
        int kk = k + 2 * hs;
        v2f b;
        b.x = W[kk * NCLS + colc] * msk;
        b.y = W[(kk + 1) * NCLS + colc] * msk;
        acc = __builtin_amdgcn_wmma_f32_16x16x4_f32(false, a, false, b,
                                                    (short)0, acc, false, false);
    }
    if (col < NCLS) {
#pragma unroll
        for (int v = 0; v < 8; ++v) {
            int m = v + 8 * hs;
            C[(size_t)(node0 + m) * NCLS + col] = acc[v];
        }
    }
}

// ---------------------------------------------------------------------------
// SpMM: out[i][f] = sum_{e in row i} vals[e] * h[col[e]][f]  (+bias, opt relu)
// Thread = (node, feature). Gathers coalesced across f; h fits in 192MB L2.
// ---------------------------------------------------------------------------
template <int F>
__global__ void spmm_kernel(const int* __restrict__ row_ptr,
                            const int* __restrict__ col,
                            const float* __restrict__ vals,
                            const float* __restrict__ h,
                            const float* __restrict__ bias,
                            float* __restrict__ out, int n, int do_relu) {
    long t = (long)blockIdx.x * blockDim.x + threadIdx.x;
    int i = (int)(t / F);
    int f = (int)(t % F);
    if (i >= n) return;
    int e0 = row_ptr[i];
    int e1 = row_ptr[i + 1];
    float acc = 0.0f;
    for (int e = e0; e < e1; ++e) {
        acc = fmaf(vals[e], h[(size_t)col[e] * F + f], acc);
    }
    acc += bias[f];
    if (do_relu) acc = fmaxf(acc, 0.0f);
    out[(size_t)i * F + f] = acc;
}

// ---------------------------------------------------------------------------
// BatchNorm (training-mode batch stats over nodes), DIM = 128 features
// ---------------------------------------------------------------------------
__global__ void zero_stats_kernel(float* __restrict__ stats) {
    stats[threadIdx.x] = 0.0f;                  // blockDim = 256 = 2*DIM
}

__global__ void bn_stats_kernel(const float* __restrict__ h,
                                float* __restrict__ stats, int n) {
    int f = threadIdx.x;                        // blockDim = DIM
    float s = 0.0f, s2 = 0.0f;
    for (int i = blockIdx.x; i < n; i += gridDim.x) {
        float v = h[(size_t)i * DIM + f];
        s += v;
        s2 += v * v;
    }
    atomicAdd(&stats[f], s);
    atomicAdd(&stats[DIM + f], s2);
}

__global__ void bn_apply_relu_kernel(float* __restrict__ h,
                                     const float* __restrict__ stats,
                                     const float* __restrict__ gamma,
                                     const float* __restrict__ beta, int n) {
    long idx = (long)blockIdx.x * blockDim.x + threadIdx.x;
    if (idx >= (long)n * DIM) return;
    int f = (int)(idx & (DIM - 1));
    float inv_n = 1.0f / (float)n;
    float mean = stats[f] * inv_n;
    float var  = stats[DIM + f] * inv_n - mean * mean;
    float v = (h[idx] - mean) * rsqrtf(var + BN_EPS) * gamma[f] + beta[f];
    h[idx] = fmaxf(v, 0.0f);
}

// ---------------------------------------------------------------------------
// Orchestration
// ---------------------------------------------------------------------------
extern "C" void kernel_launch(void* const* d_in, const int* in_sizes, int n_in,
                              void* d_out, int out_size, void* d_ws,
                              size_t ws_size, hipStream_t stream) {
    const float* x     = (const float*)d_in[0];
    const float* vals  = (const float*)d_in[1];
    const float* W1    = (const float*)d_in[2];   // (L,128,16)
    const float* b1    = (const float*)d_in[3];   // (L,16)
    const float* W2    = (const float*)d_in[4];   // (L,16,128)
    const float* b2    = (const float*)d_in[5];   // (L,128)
    const float* gamma = (const float*)d_in[6];   // (L,128)
    const float* beta  = (const float*)d_in[7];   // (L,128)
    const float* W1f   = (const float*)d_in[8];   // (128,16)
    const float* b1f   = (const float*)d_in[9];   // (16,)
    const float* W2f   = (const float*)d_in[10];  // (16,40)
    const float* b2f   = (const float*)d_in[11];  // (40,)
    const int*   row   = (const int*)d_in[12];
    const int*   col   = (const int*)d_in[13];
    float* out = (float*)d_out;

    // Workspace layout (floats)
    float* ws   = (float*)d_ws;
    float* bufA = ws;                                   // N*D  (GEMM2 output Q)
    float* bufB = bufA + (size_t)N_NODES * DIM;         // N*D  (post-SpMM / x)
    float* P    = bufB + (size_t)N_NODES * DIM;         // N*H
    float* h1   = P + (size_t)N_NODES * HID;            // N*H
    float* stats = h1 + (size_t)N_NODES * HID;          // 2*DIM
    int*   row_ptr = (int*)(stats + 2 * DIM);           // N+1 ints

    const int ntiles = N_NODES / 16;                    // 6250 (exact)

    build_row_ptr_kernel<<<(N_EDGES + 255) / 256, 256, 0, stream>>>(
        row, row_ptr, N_EDGES, N_NODES);

    const float* cur_x = x;
    for (int i = 0; i < NLAYERS; ++i) {
        gemm_d2h_wmma<<<(ntiles + 7) / 8, 256, 0, stream>>>(
            cur_x, W1 + (size_t)i * DIM * HID, P, ntiles);
        spmm_kernel<HID><<<(N_NODES * HID + 255) / 256, 256, 0, stream>>>(
            row_ptr, col, vals, P, b1 + i * HID, h1, N_NODES, 1);
        gemm_h2d_wmma<<<ntiles, 256, 0, stream>>>(
            h1, W2 + (size_t)i * HID * DIM, bufA, ntiles);
        spmm_kernel<DIM><<<(N_NODES * DIM + 255) / 256, 256, 0, stream>>>(
            row_ptr, col, vals, bufA, b2 + i * DIM, bufB, N_NODES, 0);
        zero_stats_kernel<<<1, 256, 0, stream>>>(stats);
        bn_stats_kernel<<<512, DIM, 0, stream>>>(bufB, stats, N_NODES);
        bn_apply_relu_kernel<<<(N_NODES * DIM + 255) / 256, 256, 0, stream>>>(
            bufB, stats, gamma + i * DIM, beta + i * DIM, N_NODES);
        cur_x = bufB;
    }

    // Head: GCN block with 16 -> 40 output projection
    gemm_d2h_wmma<<<(ntiles + 7) / 8, 256, 0, stream>>>(cur_x, W1f, P, ntiles);
    spmm_kernel<HID><<<(N_NODES * HID + 255) / 256, 256, 0, stream>>>(
        row_ptr, col, vals, P, b1f, h1, N_NODES, 1);
    gemm_h2c_wmma<<<ntiles, 96, 0, stream>>>(h1, W2f, bufA, ntiles);
    spmm_kernel<NCLS><<<(N_NODES * NCLS + 255) / 256, 256, 0, stream>>>(
        row_ptr, col, vals, bufA, b2f, out, N_NODES, 0);
}